// YoloLoss_11467562680721
// MI455X (gfx1250) — compile-verified
//
#include <hip/hip_runtime.h>
#include <stdint.h>

// Problem constants (from reference)
#define S7       7
#define CELLS    49          // S*S
#define DDIM     30          // B*5 + C
#define BATCH_N  16384
#define NBOX     8
#define BPB      4           // batches per block
#define THREADS  256
#define NWAVES   (THREADS / 32)
#define PRED_F4  1470        // float4s per 4-batch tile (4*1470 floats / 4)
#define LAB_F4   40          // float4s per 4-batch label tile (4*40 floats / 4)

typedef int v4i __attribute__((__vector_size__(4 * sizeof(int))));
typedef __attribute__((address_space(1))) v4i g_v4i;   // global
typedef __attribute__((address_space(3))) v4i l_v4i;   // LDS

#if defined(__has_builtin)
#  if __has_builtin(__builtin_amdgcn_global_load_async_to_lds_b128)
#    define HAVE_ASYNC_BUILTIN 1
#  endif
#  if __has_builtin(__builtin_amdgcn_s_wait_asynccnt)
#    define HAVE_WAIT_BUILTIN 1
#  endif
#endif

__device__ __forceinline__ void async_copy_b128(const float4* g, float4* l) {
#if defined(HAVE_ASYNC_BUILTIN)
    __builtin_amdgcn_global_load_async_to_lds_b128(
        (g_v4i*)(uintptr_t)g,
        (l_v4i*)(uintptr_t)l,
        /*offset=*/0, /*cpol=*/0);
#else
    asm volatile("global_load_async_to_lds_b128 %0, %1, off"
                 :: "v"((unsigned)(uintptr_t)l),
                    "v"((unsigned long long)(uintptr_t)g)
                 : "memory");
#endif
}

__device__ __forceinline__ void wait_async0() {
#if defined(HAVE_WAIT_BUILTIN)
    __builtin_amdgcn_s_wait_asynccnt(0);
#else
    asm volatile("s_wait_asynccnt 0" ::: "memory");
#endif
}

__global__ void yolo_init(float* acc) {
    if (threadIdx.x < 3) acc[threadIdx.x] = 0.0f;
}

__global__ __launch_bounds__(THREADS)
void yolo_main(const float* __restrict__ pred,
               const float* __restrict__ labels,
               float* __restrict__ acc) {
    __shared__ __align__(16) float ldsPred[BPB * 1470];   // 23520 B, DMA dest
    __shared__ __align__(16) float ldsLab[BPB * NBOX * 5];// 640 B, DMA dest
    __shared__ float tdx[BPB * CELLS], tdy[BPB * CELLS];
    __shared__ float twS[BPB * CELLS], thS[BPB * CELLS];
    __shared__ float wC[NWAVES], wL[NWAVES], wO[NWAVES];

    const int tid = threadIdx.x;
    const int blk = blockIdx.x;

    // 1) Kick off the async DMA of this block's 4-batch prediction tile.
    //    Fully unrolled: 6 outstanding per-lane b128 DMAs (4 KB each),
    //    perfectly coalesced, tracked on ASYNCcnt.
    const float4* gsrc = (const float4*)(pred + (size_t)blk * (BPB * 1470));
    float4*       ldst = (float4*)ldsPred;
    #pragma unroll
    for (int r = 0; r < 6; ++r) {
        const int i = tid + r * THREADS;
        if (r < 5 || i < PRED_F4)           // r<5 folds away at compile time
            async_copy_b128(gsrc + i, ldst + i);
    }
    // ... and the 640-B label tile (coalesced b128 instead of 4x40 scalar loads)
    if (tid < LAB_F4)
        async_copy_b128((const float4*)(labels + (size_t)blk * (BPB * NBOX * 5)) + tid,
                        ((float4*)ldsLab) + tid);

    // 2) Zero the encode arrays while the DMA is in flight.
    for (int c = tid; c < BPB * CELLS; c += THREADS) {
        tdx[c] = 0.0f; tdy[c] = 0.0f; twS[c] = 0.0f; thS[c] = 0.0f;
    }
    wait_async0();          // per-wave: all of this wave's DMAs done
    __syncthreads();        // all waves' DMAs + zeroing visible block-wide

    // 3) Encode: one thread per batch, boxes in order => last box wins,
    //    matching the JAX scatter overwrite semantics.
    if (tid < BPB) {
        const float* lab = ldsLab + tid * (NBOX * 5);
        for (int n = 0; n < NBOX; ++n) {
            float x1 = lab[n * 5 + 0], y1 = lab[n * 5 + 1];
            float x2 = lab[n * 5 + 2], y2 = lab[n * 5 + 3];
            float cx = 0.5f * (x1 + x2), cy = 0.5f * (y1 + y2);
            float fx = ceilf(cx * 7.0f) - 1.0f;
            float fy = ceilf(cy * 7.0f) - 1.0f;
            int ix = (int)fx, iy = (int)fy;
            int cell = tid * CELLS + iy * S7 + ix;
            tdx[cell] = cx * 7.0f - fx;          // dx
            tdy[cell] = cy * 7.0f - fy;          // dy
            twS[cell] = (x2 - x1) * 7.0f;        // w*S
            thS[cell] = (y2 - y1) * 7.0f;        // h*S (also the "class" slot!)
        }
    }
    __syncthreads();

    // 4) One thread per cell (196 active). Stride-30 across a wave ->
    //    32 lanes hit 32 distinct LDS banks (gcd(30,64)=2, 15 odd).
    float clsTerm = 0.0f, locTerm = 0.0f, obj = 0.0f;
    const int c = tid;
    if (c < BPB * CELLS) {
        const float* p = &ldsPred[(c / CELLS) * 1470 + (c % CELLS) * DDIM];

        // log-softmax over the 20 class logits, gathered at class_tgt
        float m = p[10];
        #pragma unroll
        for (int j = 11; j < 30; ++j) m = fmaxf(m, p[j]);
        float sum = 0.0f;
        #pragma unroll
        for (int j = 10; j < 30; ++j) sum += __expf(p[j] - m);
        float lse = m + __logf(sum);

        float hS  = thS[c];
        int   cls = (int)hS;                 // trunc(h*S); 0 for empty cells
        clsTerm = p[10 + cls] - lse;

        if (hS > 0.0f) {                     // occupied cell
            obj = 1.0f;
            float dx = tdx[c], dy = tdy[c], wS = twS[c];
            // target layout quirk: slots 5,6 are 0; slots 7,8 are dx,dy
            float a0 = p[0] - dx, a1 = p[1] - dy, a2 = p[5],      a3 = p[6];
            float b0 = p[2] - wS, b1 = p[3] - hS, b2 = p[7] - dx, b3 = p[8] - dy;
            locTerm = a0*a0 + a1*a1 + a2*a2 + a3*a3
                    + b0*b0 + b1*b1 + b2*b2 + b3*b3;
        }
    }

    // 5) wave32 shuffle reduction (no LDS memory, no barriers) ...
    #pragma unroll
    for (int off = 16; off > 0; off >>= 1) {
        clsTerm += __shfl_down(clsTerm, off, 32);
        locTerm += __shfl_down(locTerm, off, 32);
        obj     += __shfl_down(obj,     off, 32);
    }
    // ... then one cross-wave LDS stage + single barrier
    if ((tid & 31) == 0) {
        wC[tid >> 5] = clsTerm; wL[tid >> 5] = locTerm; wO[tid >> 5] = obj;
    }
    __syncthreads();
    if (tid == 0) {
        float sC = 0.0f, sL = 0.0f, sO = 0.0f;
        #pragma unroll
        for (int w = 0; w < NWAVES; ++w) { sC += wC[w]; sL += wL[w]; sO += wO[w]; }
        atomicAdd(&acc[0], sC);   // sum of gathered log-softmax terms
        atomicAdd(&acc[1], sL);   // masked localization SE
        atomicAdd(&acc[2], sO);   // n_obj
    }
}

__global__ void yolo_finalize(const float* __restrict__ acc,
                              float* __restrict__ out) {
    if (threadIdx.x == 0 && blockIdx.x == 0) {
        float class_loss = -acc[0] / (float)(BATCH_N * CELLS);
        float loc_loss   = acc[1] / (acc[2] * 4.0f);   // n_obj * (B*2)
        out[0] = class_loss + 5.0f * loc_loss;         // + L_COORD * loc
    }
}

extern "C" void kernel_launch(void* const* d_in, const int* in_sizes, int n_in,
                              void* d_out, int out_size, void* d_ws, size_t ws_size,
                              hipStream_t stream) {
    const float* pred   = (const float*)d_in[0];   // [16384, 1470] f32
    const float* labels = (const float*)d_in[1];   // [16384, 8, 5] f32
    float*       acc    = (float*)d_ws;            // 3 accumulators
    float*       out    = (float*)d_out;           // scalar

    yolo_init<<<1, 32, 0, stream>>>(acc);
    yolo_main<<<BATCH_N / BPB, THREADS, 0, stream>>>(pred, labels, acc);
    yolo_finalize<<<1, 32, 0, stream>>>(acc, out);
}